// NNGuide_65584150610439
// MI455X (gfx1250) — compile-verified
//
#include <hip/hip_runtime.h>
#include <math.h>

// ---------------- problem constants ----------------
#define B_Q   2048
#define CCLS  1000
#define DIM   512
#define NBANK 100000
#define KTOP  10

// ---------------- tiling ----------------
#define QSUP   64                 // queries per block (4 waves x 16)
#define NT     2                  // 16-row bank tiles per pass (32 bank rows)
#define PASSES 125                // passes per chunk
#define CHUNK  (NT * 16 * PASSES) // 4000 bank rows per chunk
#define NCHUNK 25                 // 25*4000 = 100000 exactly -> no tail masking
#define NSUP   (B_Q / QSUP)       // 32 query supertiles

// LDS layout (floats), padded strides for conflict-free access
#define LDA        516            // 516 % 64 == 4 -> conflict-free b64 fragment loads
#define SCR_STRIDE 17             // 17*q % 64 distinct for q=0..15
#define SCR_PER_WAVE (NT * 16 * SCR_STRIDE)
#define BUF_FLOATS (NT * 16 * LDA)              // one B buffer: 32 rows
#define OFF_A 0
#define OFF_B (QSUP * LDA)                      // double-buffered B starts here
#define OFF_S (QSUP * LDA + 2 * BUF_FLOATS)
#define LDS_FLOATS (QSUP * LDA + 2 * BUF_FLOATS + 4 * SCR_PER_WAVE)
// = 33024 + 33024 + 2176 = 68224 floats = 272,896 B (< 320 KB/WGP)

typedef __attribute__((ext_vector_type(2))) float v2f;
typedef __attribute__((ext_vector_type(8))) float v8f;

__device__ __forceinline__ v8f wmma_f32_16x16x4(v2f a, v2f b, v8f c) {
    // D = A(16x4,f32) x B(4x16,f32) + C(16x16,f32), wave32
    return __builtin_amdgcn_wmma_f32_16x16x4_f32(
        /*neg_a=*/false, a, /*neg_b=*/false, b,
        /*c_mod=*/(short)0, c, /*reuse_a=*/false, /*reuse_b=*/false);
}

// Async VGPR-free copy: 16 bytes per lane, global -> LDS, tracked by ASYNCcnt.
__device__ __forceinline__ void async_load_b128(unsigned ldsByteAddr, const float* gp) {
    asm volatile("global_load_async_to_lds_b128 %0, %1, off"
                 :: "v"(ldsByteAddr),
                    "v"((unsigned long long)(uintptr_t)gp)
                 : "memory");
}

// branchless sorted top-K insert (descending)
__device__ __forceinline__ void topk_insert(float (&best)[KTOP], float v) {
#pragma unroll
    for (int j = 0; j < KTOP; ++j) {
        float mx = fmaxf(best[j], v);
        v = fminf(best[j], v);
        best[j] = mx;
    }
}

// ---------------- kernel 1: logsumexp over classes ----------------
__global__ void __launch_bounds__(256)
conf_kernel(const float* __restrict__ logits, float* __restrict__ confs) {
    const int wave = threadIdx.x >> 5;
    const int lane = threadIdx.x & 31;
    const int row  = blockIdx.x * 8 + wave;
    if (row >= B_Q) return;
    const float* rp = logits + (size_t)row * CCLS;

    float m = -INFINITY;
    for (int i = lane; i < CCLS; i += 32) m = fmaxf(m, rp[i]);
#pragma unroll
    for (int off = 16; off > 0; off >>= 1) m = fmaxf(m, __shfl_xor(m, off, 32));

    float s = 0.f;
    for (int i = lane; i < CCLS; i += 32) s += __expf(rp[i] - m);
#pragma unroll
    for (int off = 16; off > 0; off >>= 1) s += __shfl_xor(s, off, 32);

    if (lane == 0) confs[row] = m + __logf(s);
}

// ---------------- kernel 2: WMMA GEMM + streaming per-chunk top-10 ----------------
// grid = (NSUP, NCHUNK): blockIdx.x = query supertile (fast dim so the 32
// concurrent blocks stream the same ~8MB bank chunk out of the 192MB L2),
// blockIdx.y = bank chunk. B-tiles are double-buffered via async-to-LDS copies
// so staging overlaps the WMMA stream.
__global__ void __launch_bounds__(128, 1)
sim_topk_kernel(const float* __restrict__ feats, const float* __restrict__ bank,
                float* __restrict__ cand) {
    extern __shared__ float smem[];
    float* As = smem + OFF_A;
    float* Bs = smem + OFF_B;
    float* Ss = smem + OFF_S;
    // Flat->LDS: low 32 bits of a flat shared pointer are the LDS byte address.
    const unsigned ldsBase = (unsigned)(uintptr_t)smem;

    const int tid  = threadIdx.x;
    const int wave = tid >> 5;
    const int lane = tid & 31;
    const int half = lane >> 4;     // 0: lanes 0-15, 1: lanes 16-31
    const int mn   = lane & 15;     // M for A-frag, N for B-frag, N for C/D
    const int qbase = blockIdx.x * QSUP;
    const long cbase = (long)blockIdx.y * CHUNK;
    const int chunk = blockIdx.y;

    // Stage A supertile: 64 x 512 f32, one float4 per thread per row.
    for (int j = 0; j < QSUP; ++j) {
        const float4 v = *(const float4*)(feats + (size_t)(qbase + j) * DIM + tid * 4);
        *(float4*)(As + j * LDA + tid * 4) = v;
    }

    float best[KTOP];
#pragma unroll
    for (int j = 0; j < KTOP; ++j) best[j] = -INFINITY;

    float* scr = Ss + wave * SCR_PER_WAVE;
    const float* aP = As + (wave * 16 + mn) * LDA + 2 * half;

    // LDS/global bases for this thread's staging slice (thread t owns float4
    // column t of every staged row).
    const unsigned dstT = ldsBase + 4u * (unsigned)(OFF_B + tid * 4);
    const float* gT = bank + (size_t)tid * 4;

    // Prologue: kick off pass 0 into buffer 0 (32 async b128 per wave).
    {
        const float* g0 = gT + (size_t)cbase * DIM;
#pragma unroll 4
        for (int j = 0; j < NT * 16; ++j)
            async_load_b128(dstT + (unsigned)(j * LDA * 4), g0 + (size_t)j * DIM);
    }

    for (int pass = 0; pass < PASSES; ++pass) {
        const int cur = pass & 1;
        const long rbase = cbase + (long)pass * (NT * 16);

        // Issue next pass into the other buffer, then wait for current pass.
        // Async loads complete in order per wave: after issuing 32 more,
        // ASYNCcnt<=32 implies the current pass's 32 have landed.
        if (pass + 1 < PASSES) {
            const float* gn = gT + (size_t)(rbase + NT * 16) * DIM;
            const unsigned dn = dstT + (unsigned)((1 - cur) * BUF_FLOATS * 4);
#pragma unroll 4
            for (int j = 0; j < NT * 16; ++j)
                async_load_b128(dn + (unsigned)(j * LDA * 4), gn + (size_t)j * DIM);
            asm volatile("s_wait_asynccnt 0x20" ::: "memory");
        } else {
            asm volatile("s_wait_asynccnt 0x0" ::: "memory");
        }
        __syncthreads();   // all waves' column slabs (and pass-0 A stage) visible

        // 2 independent 16x16 accumulators per wave.
        v8f acc[NT];
#pragma unroll
        for (int t = 0; t < NT; ++t) acc[t] = (v8f){0, 0, 0, 0, 0, 0, 0, 0};

        const float* bP = Bs + cur * BUF_FLOATS + mn * LDA + 2 * half;
#pragma unroll 4
        for (int kb = 0; kb < DIM / 4; ++kb) {
            const v2f a = *(const v2f*)(aP + kb * 4);
#pragma unroll
            for (int t = 0; t < NT; ++t) {
                const v2f b = *(const v2f*)(bP + t * 16 * LDA + kb * 4);
                acc[t] = wmma_f32_16x16x4(a, b, acc[t]);
            }
        }

        // Spill sims to conflict-free LDS scratch. C/D layout: VGPR j holds
        // M = j + 8*half, N = mn.
#pragma unroll
        for (int t = 0; t < NT; ++t) {
#pragma unroll
            for (int j = 0; j < 8; ++j) {
                scr[t * 16 * SCR_STRIDE + (j + 8 * half) * SCR_STRIDE + mn] = acc[t][j];
            }
        }
        asm volatile("s_wait_dscnt 0" ::: "memory");   // same-wave DS RAW fence
        __builtin_amdgcn_wave_barrier();

        // Lanes 0-15: each owns query row `lane`; scan 32 new sims into top-10.
        if (lane < 16) {
#pragma unroll 1
            for (int t = 0; t < NT; ++t) {
                const float* row = scr + t * 16 * SCR_STRIDE + lane * SCR_STRIDE;
                for (int n = 0; n < 16; ++n) topk_insert(best, row[n]);
            }
        }
        // Barrier before any wave issues async writes into the buffer that
        // other waves may still be reading this pass.
        __syncthreads();
    }

    // Emit per-(query, chunk) top-10 candidates.
    if (lane < 16) {
        const int q = qbase + wave * 16 + lane;
        float* cp = cand + (size_t)q * (NCHUNK * KTOP) + chunk * KTOP;
#pragma unroll
        for (int j = 0; j < KTOP; ++j) cp[j] = best[j];
    }
}

// ---------------- kernel 3: merge candidates, mean * conf ----------------
__global__ void __launch_bounds__(256)
merge_kernel(const float* __restrict__ cand, const float* __restrict__ confs,
             float* __restrict__ out) {
    const int q = blockIdx.x * blockDim.x + threadIdx.x;
    if (q >= B_Q) return;
    const float* cp = cand + (size_t)q * (NCHUNK * KTOP);
    float best[KTOP];
#pragma unroll
    for (int j = 0; j < KTOP; ++j) best[j] = -INFINITY;
    for (int i = 0; i < NCHUNK * KTOP; ++i) topk_insert(best, cp[i]);
    float s = 0.f;
#pragma unroll
    for (int j = 0; j < KTOP; ++j) s += best[j];
    out[q] = (s * (1.0f / KTOP)) * confs[q];
}

// ---------------- launcher ----------------
extern "C" void kernel_launch(void* const* d_in, const int* in_sizes, int n_in,
                              void* d_out, int out_size, void* d_ws, size_t ws_size,
                              hipStream_t stream) {
    const float* logits = (const float*)d_in[0];   // [2048,1000]
    const float* feats  = (const float*)d_in[1];   // [2048,512]
    const float* bank   = (const float*)d_in[2];   // [100000,512]
    float* out = (float*)d_out;                    // [2048] f32

    float* confs = (float*)d_ws;                   // 2048 floats
    float* cand  = confs + B_Q;                    // 2048 * 250 floats (~2MB)

    conf_kernel<<<B_Q / 8, 256, 0, stream>>>(logits, confs);
    sim_topk_kernel<<<dim3(NSUP, NCHUNK), 128, LDS_FLOATS * sizeof(float), stream>>>(
        feats, bank, cand);
    merge_kernel<<<(B_Q + 255) / 256, 256, 0, stream>>>(cand, confs, out);
}